// MossTTSRealtime_76338748719417
// MI455X (gfx1250) — compile-verified
//
#include <hip/hip_runtime.h>
#include <math.h>

// ---------------- model constants (from reference) ----------------
#define LAY   6
#define BB    32
#define HH    1024
#define NHH   16
#define KVHH  4
#define DD    64
#define INTER 4096
#define SS    16
#define VOC   1027
#define EPSF  1e-6f
#define NSPLIT 8                       // split-K factor (deterministic 2-stage)

typedef __attribute__((ext_vector_type(2))) float v2f;
typedef __attribute__((ext_vector_type(8))) float v8f;

__device__ __forceinline__ v8f wmma_f32_k4(v2f a, v2f b, v8f c) {
  // V_WMMA_F32_16X16X4_F32: D = A(16x4) x B(4x16) + C(16x16), fp32, wave32
  return __builtin_amdgcn_wmma_f32_16x16x4_f32(false, a, false, b, (short)0, c,
                                               false, false);
}

// =================================================================
// RMSNorm: one block per row (B=32 rows), 256 threads
// =================================================================
__global__ void rmsnorm_kernel(const float* __restrict__ X,
                               const float* __restrict__ w,
                               float* __restrict__ Y, int n) {
  __shared__ float red[256];
  const int row = blockIdx.x;
  const float* x = X + (size_t)row * n;
  float s = 0.f;
  for (int i = threadIdx.x; i < n; i += 256) { float t = x[i]; s += t * t; }
  red[threadIdx.x] = s;
  __syncthreads();
  for (int off = 128; off > 0; off >>= 1) {
    if (threadIdx.x < off) red[threadIdx.x] += red[threadIdx.x + off];
    __syncthreads();
  }
  const float rs = rsqrtf(red[0] / (float)n + EPSF);
  for (int i = threadIdx.x; i < n; i += 256)
    Y[(size_t)row * n + i] = w[i] * x[i] * rs;
}

// =================================================================
// Split-K GEMM partial: P[split](32 x N) = X(32 x CHUNK) @ W(CHUNK x N)
// Compile-time K/N/CHUNK -> immediate-offset loads, strength-reduced bases.
// block = 128 threads = 4 waves; each wave: 2 N-tiles (32 cols) x full M=32.
// A fragments shared across both B tiles: 4 WMMAs per K-step of 4.
// =================================================================
template <int K, int N, int CHUNK>
__global__ __launch_bounds__(128) void gemm32_partial(
    const float* __restrict__ X, const float* __restrict__ W,
    float* __restrict__ P) {
  const int lane = threadIdx.x & 31;
  const int wv   = threadIdx.x >> 5;
  const int n0   = (blockIdx.x * 4 + wv) * 32;   // 2 tiles of 16 cols per wave
  if (n0 >= N) return;

  const int kbeg = blockIdx.y * CHUNK;

  const int half = lane >> 4;                    // 0 -> K rows {0,1}, 1 -> {2,3}
  const int l16  = lane & 15;
  const int col0 = n0 + l16;
  const int col1 = n0 + 16 + l16;
  constexpr bool EDGE = (N % 32) != 0;
  const bool cok0 = !EDGE || (col0 < N);
  const bool cok1 = !EDGE || (col1 < N);

  v8f c00 = {}, c01 = {}, c10 = {}, c11 = {};    // [ntile][mtile]

  const float* xa0 = X + l16 * K + 2 * half + kbeg;          // rows 0..15
  const float* xa1 = X + (16 + l16) * K + 2 * half + kbeg;   // rows 16..31
  const float* wb0 = W + (size_t)(kbeg + 2 * half) * N + (cok0 ? col0 : n0);
  const float* wb1 = W + (size_t)(kbeg + 2 * half) * N + (cok1 ? col1 : n0);

  for (int kk = 0; kk < CHUNK; kk += 16) {
    // stream-prefetch the weight tile ~48 K-rows ahead (speculative, OOB-safe)
    __builtin_prefetch(wb0 + (size_t)(kk + 48) * N, 0, 1);
    __builtin_prefetch(wb1 + (size_t)(kk + 48) * N, 0, 1);
#pragma unroll
    for (int k0 = kk; k0 < kk + 16; k0 += 4) {
      v2f a0 = *(const v2f*)(xa0 + k0);
      v2f a1 = *(const v2f*)(xa1 + k0);
      float t00 = wb0[(size_t)k0 * N];
      float t01 = wb0[(size_t)(k0 + 1) * N];
      float t10 = wb1[(size_t)k0 * N];
      float t11 = wb1[(size_t)(k0 + 1) * N];
      if (EDGE) {
        if (!cok0) { t00 = 0.f; t01 = 0.f; }
        if (!cok1) { t10 = 0.f; t11 = 0.f; }
      }
      v2f b0, b1;
      b0[0] = t00; b0[1] = t01;
      b1[0] = t10; b1[1] = t11;
      c00 = wmma_f32_k4(a0, b0, c00);
      c01 = wmma_f32_k4(a1, b0, c01);
      c10 = wmma_f32_k4(a0, b1, c10);
      c11 = wmma_f32_k4(a1, b1, c11);
    }
  }

  // C layout: VGPR r -> M=r (lanes 0-15), M=8+r (lanes 16-31)
  float* p = P + (size_t)blockIdx.y * 32 * N;
#pragma unroll
  for (int r = 0; r < 8; ++r) {
    const int row0 = r + 8 * half;
    const int row1 = 16 + r + 8 * half;
    if (cok0) {
      p[(size_t)row0 * N + col0] = c00[r];
      p[(size_t)row1 * N + col0] = c01[r];
    }
    if (cok1) {
      p[(size_t)row0 * N + col1] = c10[r];
      p[(size_t)row1 * N + col1] = c11[r];
    }
  }
}

// =================================================================
// Reduce split-K partials (float4 vectorized); optional residual += into Y
// =================================================================
__global__ void reduce_add(const float* __restrict__ P, float* __restrict__ Y,
                           int total4, int nsplit, int addRes) {
  const int i = blockIdx.x * 256 + threadIdx.x;
  if (i >= total4) return;
  const float4* p4 = (const float4*)P;
  float4 s = make_float4(0.f, 0.f, 0.f, 0.f);
  for (int sp = 0; sp < nsplit; ++sp) {
    float4 t = p4[(size_t)sp * total4 + i];
    s.x += t.x; s.y += t.y; s.z += t.z; s.w += t.w;
  }
  float4* y4 = (float4*)Y;
  if (addRes) {
    float4 o = y4[i];
    s.x += o.x; s.y += o.y; s.z += o.z; s.w += o.w;
  }
  y4[i] = s;
}

// Reduce gate/up partials and apply SwiGLU: A = silu(g) * u
__global__ void reduce_swiglu(const float* __restrict__ Pg,
                              const float* __restrict__ Pu,
                              float* __restrict__ A, int total4, int nsplit) {
  const int i = blockIdx.x * 256 + threadIdx.x;
  if (i >= total4) return;
  const float4* pg4 = (const float4*)Pg;
  const float4* pu4 = (const float4*)Pu;
  float4 g = make_float4(0.f, 0.f, 0.f, 0.f);
  float4 u = make_float4(0.f, 0.f, 0.f, 0.f);
  for (int sp = 0; sp < nsplit; ++sp) {
    float4 tg = pg4[(size_t)sp * total4 + i];
    float4 tu = pu4[(size_t)sp * total4 + i];
    g.x += tg.x; g.y += tg.y; g.z += tg.z; g.w += tg.w;
    u.x += tu.x; u.y += tu.y; u.z += tu.z; u.w += tu.w;
  }
  float4 r;
  r.x = g.x / (1.f + expf(-g.x)) * u.x;
  r.y = g.y / (1.f + expf(-g.y)) * u.y;
  r.z = g.z / (1.f + expf(-g.z)) * u.z;
  r.w = g.w / (1.f + expf(-g.w)) * u.w;
  ((float4*)A)[i] = r;
}

// =================================================================
// Attention for one (batch, q-head): q/k RMSNorm + RoPE + cache attn.
// grid = B*NH, block = 64 (=D). KV cache inputs are read-only; the
// new k/v at step_idx are substituted from LDS.
// =================================================================
__global__ void attn_kernel(const float* __restrict__ qb,
                            const float* __restrict__ kb,
                            const float* __restrict__ vb,
                            const float* __restrict__ qnw,
                            const float* __restrict__ knw,
                            const float* __restrict__ kc,
                            const float* __restrict__ vc,
                            const int* __restrict__ stepp,
                            float* __restrict__ ob) {
  const int bh  = blockIdx.x;
  const int b   = bh >> 4;          // / NH
  const int h   = bh & 15;          // % NH
  const int kvh = h >> 2;           // / (NH/KVH)
  const int d   = threadIdx.x;      // 0..63
  const int step = stepp[0];

  __shared__ float red[64];
  __shared__ float qs[64], ks[64], vs[64];
  __shared__ float pr[SS];

  float q = qb[((size_t)b * NHH  + h)   * DD + d];
  float k = kb[((size_t)b * KVHH + kvh) * DD + d];
  float v = vb[((size_t)b * KVHH + kvh) * DD + d];

  // RMS(q) over D
  red[d] = q * q; __syncthreads();
  for (int off = 32; off > 0; off >>= 1) { if (d < off) red[d] += red[d + off]; __syncthreads(); }
  const float qr = rsqrtf(red[0] * (1.f / DD) + EPSF);
  __syncthreads();
  // RMS(k) over D
  red[d] = k * k; __syncthreads();
  for (int off = 32; off > 0; off >>= 1) { if (d < off) red[d] += red[d + off]; __syncthreads(); }
  const float kr = rsqrtf(red[0] * (1.f / DD) + EPSF);
  __syncthreads();

  q = q * qr * qnw[d];
  k = k * kr * knw[d];
  qs[d] = q; ks[d] = k; vs[d] = v;
  __syncthreads();

  // RoPE: inv_freq[i] = THETA^(-i/32), emb = concat(freqs, freqs)
  const int   i    = d & 31;
  const float invf = powf(1000000.f, -(float)i / 32.f);
  const float ang  = (float)step * invf;
  const float cs = cosf(ang), sn = sinf(ang);
  const float qrot = (d < 32) ? -qs[d + 32] : qs[d - 32];
  const float krot = (d < 32) ? -ks[d + 32] : ks[d - 32];
  q = q * cs + qrot * sn;
  k = k * cs + krot * sn;
  __syncthreads();
  qs[d] = q; ks[d] = k;
  __syncthreads();

  // scores over S=16 cache positions (new k substituted at step_idx)
  if (d < SS) {
    const int j = d;
    const float* kj = kc + (((size_t)b * KVHH + kvh) * SS + j) * DD;
    float acc = 0.f;
    if (j == step) { for (int t = 0; t < DD; ++t) acc += qs[t] * ks[t]; }
    else           { for (int t = 0; t < DD; ++t) acc += qs[t] * kj[t]; }
    pr[j] = acc * 0.125f + ((j <= step) ? 0.f : -1e9f);
  }
  __syncthreads();

  if (d == 0) {                    // softmax over 16 values
    float m = pr[0];
    for (int j = 1; j < SS; ++j) m = fmaxf(m, pr[j]);
    float sum = 0.f;
    for (int j = 0; j < SS; ++j) { float e = expf(pr[j] - m); pr[j] = e; sum += e; }
    const float inv = 1.f / sum;
    for (int j = 0; j < SS; ++j) pr[j] *= inv;
  }
  __syncthreads();

  float out = 0.f;
  for (int j = 0; j < SS; ++j) {
    const float* vj = vc + (((size_t)b * KVHH + kvh) * SS + j) * DD;
    out += pr[j] * ((j == step) ? vs[d] : vj[d]);
  }
  ob[((size_t)b * NHH + h) * DD + d] = out;
}

// =================================================================
// Host-side orchestration
// =================================================================
template <int K, int N, int CHUNK>
static void launch_gemm(const float* X, const float* W, float* Y, float* P,
                        int addRes, hipStream_t stream) {
  dim3 grid((N + 127) / 128, K / CHUNK);
  gemm32_partial<K, N, CHUNK><<<grid, 128, 0, stream>>>(X, W, P);
  const int total4 = (32 * N) / 4;
  reduce_add<<<(total4 + 255) / 256, 256, 0, stream>>>(P, Y, total4, K / CHUNK,
                                                       addRes);
}

extern "C" void kernel_launch(void* const* d_in, const int* in_sizes, int n_in,
                              void* d_out, int out_size, void* d_ws, size_t ws_size,
                              hipStream_t stream) {
  const float* bh   = (const float*)d_in[0];
  const float* Wq   = (const float*)d_in[1];
  const float* Wk   = (const float*)d_in[2];
  const float* Wv   = (const float*)d_in[3];
  const float* Wo   = (const float*)d_in[4];
  const float* qnw  = (const float*)d_in[5];
  const float* knw  = (const float*)d_in[6];
  const float* ln1  = (const float*)d_in[7];
  const float* ln2  = (const float*)d_in[8];
  const float* Wg   = (const float*)d_in[9];
  const float* Wu   = (const float*)d_in[10];
  const float* Wd   = (const float*)d_in[11];
  const float* fnw  = (const float*)d_in[12];
  const float* lmh  = (const float*)d_in[13];
  const float* kc   = (const float*)d_in[14];
  const float* vc   = (const float*)d_in[15];
  const int*   step = (const int*)d_in[16];
  float* out = (float*)d_out;

  // workspace layout (floats); total ~9.5 MB
  float* ws = (float*)d_ws;
  float* h  = ws;                          // 32*1024
  float* xn = h  + 32 * HH;                // 32*1024
  float* qB = xn + 32 * HH;                // 32*1024
  float* kB = qB + 32 * NHH * DD;          // 32*256
  float* vB = kB + 32 * KVHH * DD;         // 32*256
  float* oB = vB + 32 * KVHH * DD;         // 32*1024
  float* aB = oB + 32 * NHH * DD;          // 32*4096
  float* P0 = aB + 32 * INTER;             // NSPLIT*32*4096
  float* P1 = P0 + (size_t)NSPLIT * 32 * INTER;

  hipMemcpyAsync(h, bh, (size_t)32 * HH * sizeof(float),
                 hipMemcpyDeviceToDevice, stream);

  for (int l = 0; l < LAY; ++l) {
    rmsnorm_kernel<<<32, 256, 0, stream>>>(h, ln1 + (size_t)l * HH, xn, HH);

    launch_gemm<HH, NHH * DD, HH / NSPLIT>(
        xn, Wq + (size_t)l * HH * NHH * DD, qB, P0, 0, stream);
    launch_gemm<HH, KVHH * DD, HH / NSPLIT>(
        xn, Wk + (size_t)l * HH * KVHH * DD, kB, P0, 0, stream);
    launch_gemm<HH, KVHH * DD, HH / NSPLIT>(
        xn, Wv + (size_t)l * HH * KVHH * DD, vB, P0, 0, stream);

    attn_kernel<<<BB * NHH, 64, 0, stream>>>(
        qB, kB, vB, qnw + (size_t)l * DD, knw + (size_t)l * DD,
        kc + (size_t)l * BB * KVHH * SS * DD,
        vc + (size_t)l * BB * KVHH * SS * DD, step, oB);

    launch_gemm<NHH * DD, HH, HH / NSPLIT>(
        oB, Wo + (size_t)l * NHH * DD * HH, h, P0, 1, stream);

    rmsnorm_kernel<<<32, 256, 0, stream>>>(h, ln2 + (size_t)l * HH, xn, HH);

    {
      dim3 grid((INTER + 127) / 128, NSPLIT);
      gemm32_partial<HH, INTER, HH / NSPLIT>
          <<<grid, 128, 0, stream>>>(xn, Wg + (size_t)l * HH * INTER, P0);
      gemm32_partial<HH, INTER, HH / NSPLIT>
          <<<grid, 128, 0, stream>>>(xn, Wu + (size_t)l * HH * INTER, P1);
      const int total4 = (32 * INTER) / 4;
      reduce_swiglu<<<(total4 + 255) / 256, 256, 0, stream>>>(P0, P1, aB,
                                                              total4, NSPLIT);
    }

    launch_gemm<INTER, HH, INTER / NSPLIT>(
        aB, Wd + (size_t)l * INTER * HH, h, P0, 1, stream);
  }

  rmsnorm_kernel<<<32, 256, 0, stream>>>(h, fnw, xn, HH);
  launch_gemm<HH, VOC, HH / NSPLIT>(xn, lmh, out, P0, 0, stream);
}